// CCMetrics_23398981829269
// MI455X (gfx1250) — compile-verified
//
#include <hip/hip_runtime.h>
#include <hip/hip_bf16.h>
#include <stdint.h>

#define NSEG 65          // component ids 0..64 (0 = background)
#define EPSV 1e-5f

// ---------------- CDNA5 async global->LDS support (guarded) ----------------
#if defined(__HIP_DEVICE_COMPILE__) && defined(__gfx1250__)
#  if __has_builtin(__builtin_amdgcn_global_load_async_to_lds_b128)
#    define HAVE_ASYNC_LDS 1
#  endif
#  if __has_builtin(__builtin_amdgcn_s_wait_asynccnt)
#    define HAVE_WAIT_ASYNC 1
#  endif
#endif

#ifdef HAVE_ASYNC_LDS
typedef int v4i_t __attribute__((ext_vector_type(4)));
typedef __attribute__((address_space(1))) v4i_t gm_v4i;
typedef __attribute__((address_space(3))) v4i_t lds_v4i;
__device__ __forceinline__ void async_copy16(const float* g, float* l) {
    // 16-byte async copy, tracked by ASYNCcnt; EXEC-predicated per lane.
    __builtin_amdgcn_global_load_async_to_lds_b128((gm_v4i*)g, (lds_v4i*)l, 0, 0);
}
#endif

__device__ __forceinline__ void wait_async0() {
#if defined(HAVE_WAIT_ASYNC)
    __builtin_amdgcn_s_wait_asynccnt(0);
#elif defined(__HIP_DEVICE_COMPILE__) && defined(__gfx1250__)
    asm volatile("s_wait_asynccnt 0" ::: "memory");
#endif
}

// ---------------- kernel 0: zero the global bins ----------------
__global__ void cc_zero(float* __restrict__ bins, int n) {
    int i = blockIdx.x * blockDim.x + threadIdx.x;
    if (i < n) bins[i] = 0.0f;
}

// ---------------- kernel 1: segmented reduction ----------------
// bins layout: float bins[B][NSEG][2]  (0: inter-sum, 1: count)
__global__ __launch_bounds__(256) void cc_reduce(
    const float* __restrict__ pred,   // [B,2,N]
    const int*   __restrict__ lbl,    // [B,N]   values 0/1
    const int*   __restrict__ seg,    // [B,N]   values 0..64
    float*       __restrict__ bins,   // [B,NSEG,2]
    int N)                            // voxels per sample
{
    const int b = blockIdx.y;
    const float* c0 = pred + (size_t)b * 2u * (size_t)N;
    const float* c1 = c0 + N;
    const int*   yb = lbl + (size_t)b * (size_t)N;
    const int*   sb = seg + (size_t)b * (size_t)N;
    float*       gb = bins + b * NSEG * 2;

    __shared__ float lbins[NSEG * 2];
    __shared__ float s0[1024];        // async staging: channel 0
    __shared__ float s1[1024];        // async staging: channel 1

    const int tid = threadIdx.x;
    for (int i = tid; i < NSEG * 2; i += 256) lbins[i] = 0.0f;
    __syncthreads();

    const int stride = gridDim.x * 1024;
    for (int base = blockIdx.x * 1024; base < N; base += stride) {
        const int idx = base + tid * 4;
        if (idx + 4 <= N) {
            const int4 yv4 = *(const int4*)(yb + idx);
            const int4 sv4 = *(const int4*)(sb + idx);
            float x0[4], x1[4];
#ifdef HAVE_ASYNC_LDS
            async_copy16(c0 + idx, &s0[tid * 4]);
            async_copy16(c1 + idx, &s1[tid * 4]);
            wait_async0();
            {
                float4 a = *(const float4*)(&s0[tid * 4]);
                float4 c = *(const float4*)(&s1[tid * 4]);
                x0[0]=a.x; x0[1]=a.y; x0[2]=a.z; x0[3]=a.w;
                x1[0]=c.x; x1[1]=c.y; x1[2]=c.z; x1[3]=c.w;
            }
#else
            {
                float4 a = *(const float4*)(c0 + idx);
                float4 c = *(const float4*)(c1 + idx);
                x0[0]=a.x; x0[1]=a.y; x0[2]=a.z; x0[3]=a.w;
                x1[0]=c.x; x1[1]=c.y; x1[2]=c.z; x1[3]=c.w;
            }
#endif
            const int yv[4] = {yv4.x, yv4.y, yv4.z, yv4.w};
            const int sv[4] = {sv4.x, sv4.y, sv4.z, sv4.w};
#pragma unroll
            for (int j = 0; j < 4; ++j) {
                // p(true class) = sigmoid(x_true - x_other)
                float d = (yv[j] != 0) ? (x0[j] - x1[j]) : (x1[j] - x0[j]);
                float p = __fdividef(1.0f, 1.0f + __expf(d));
                int s = sv[j];
                s = (s < 0) ? 0 : (s > NSEG - 1 ? NSEG - 1 : s);
                atomicAdd(&lbins[s * 2 + 0], p);      // ds_add_f32
                atomicAdd(&lbins[s * 2 + 1], 1.0f);   // ds_add_f32
            }
        } else {
            for (int j = 0; j < 4; ++j) {
                int k = idx + j;
                if (k < N) {
                    float a = c0[k], c = c1[k];
                    float d = (yb[k] != 0) ? (a - c) : (c - a);
                    float p = __fdividef(1.0f, 1.0f + __expf(d));
                    int s = sb[k];
                    s = (s < 0) ? 0 : (s > NSEG - 1 ? NSEG - 1 : s);
                    atomicAdd(&lbins[s * 2 + 0], p);
                    atomicAdd(&lbins[s * 2 + 1], 1.0f);
                }
            }
        }
    }

    __syncthreads();
    for (int i = tid; i < NSEG * 2; i += 256) {
        float v = lbins[i];
        if (v != 0.0f) atomicAdd(&gb[i], v);          // global_atomic_add_f32
    }
}

// ---------------- kernel 2: finalize dice + means ----------------
__global__ __launch_bounds__(128) void cc_final(
    const float* __restrict__ bins,   // [2][NSEG][2]
    float* __restrict__ out)
{
    __shared__ float ssum[2];
    __shared__ float scnt[2];
    const int t = threadIdx.x;
    if (t < 2) { ssum[t] = 0.0f; scnt[t] = 0.0f; }
    __syncthreads();
    if (t < 128) {                     // b = t>>6, label = (t&63)+1 (skip bg 0)
        const int b = t >> 6;
        const int s = (t & 63) + 1;
        const float I = bins[(b * NSEG + s) * 2 + 0];
        const float C = bins[(b * NSEG + s) * 2 + 1];
        if (C > 0.0f) {
            float dice = (2.0f * I + EPSV) / (2.0f * C + EPSV);
            atomicAdd(&ssum[b], dice);
            atomicAdd(&scnt[b], 1.0f);
        }
    }
    __syncthreads();
    if (t == 0) {
        float acc = 0.0f;
        for (int b = 0; b < 2; ++b)
            acc += ssum[b] / fmaxf(scnt[b], 1.0f);
        out[0] = acc * 0.5f;
    }
}

// ---------------- host entry ----------------
extern "C" void kernel_launch(void* const* d_in, const int* in_sizes, int n_in,
                              void* d_out, int out_size, void* d_ws, size_t ws_size,
                              hipStream_t stream) {
    (void)n_in; (void)out_size; (void)ws_size;
    const float* pred = (const float*)d_in[0];   // y_pred fp32 [B,2,H,W,D]
    const int*   y    = (const int*)d_in[1];     // labels int32 [B,1,H,W,D]
    const int*   seg  = (const int*)d_in[2];     // voronoi int32 [B,H,W,D]
    float* bins = (float*)d_ws;                  // [B][NSEG][2] partials
    float* out  = (float*)d_out;

    const int B = 2;
    const int N = in_sizes[2] / B;               // voxels per sample (H*W*D)
    const int nbins = B * NSEG * 2;

    cc_zero<<<1, 256, 0, stream>>>(bins, nbins);

    int chunks = (N + 1023) / 1024;
    int gx = chunks < 512 ? chunks : 512;        // grid-stride; bounds flush contention
    dim3 grid(gx, B);
    cc_reduce<<<grid, 256, 0, stream>>>(pred, y, seg, bins, N);

    cc_final<<<1, 128, 0, stream>>>(bins, out);
}